// GAT_44487271252524
// MI455X (gfx1250) — compile-verified
//
#include <hip/hip_runtime.h>
#include <hip/hip_bf16.h>

typedef __attribute__((ext_vector_type(2))) float v2f;
typedef __attribute__((ext_vector_type(8))) float v8f;

// ---------------- constants ----------------
static constexpr int NB    = 64;      // B
static constexpr int PASTN = 128;
static constexpr int FUTN  = 64;
static constexpr int FDIM  = 51;      // F
static constexpr int FM1   = 50;      // F-1
static constexpr int HEADS = 4;
static constexpr int HIDD  = 64;
static constexpr int NN1   = NB * PASTN;   // 8192
static constexpr int NN2   = NB * FUTN;    // 4096
static constexpr int EE1   = NN1 * 16;     // 131072
static constexpr int EE2   = NN2 * 16;     // 65536
static constexpr int ADIM  = PASTN + FUTN; // 192
static constexpr int XS    = 52;           // padded feature stride (even, %4==0)
static constexpr int NP1   = 208;          // padded N for h1 (200 -> 13 tiles)

static __device__ __forceinline__ float leaky02(float x) { return x > 0.f ? x : 0.2f * x; }

static __device__ __forceinline__ void atomicMaxF(float* addr, float v) {
  if (v >= 0.f) atomicMax((int*)addr, __float_as_int(v));
  else          atomicMin((unsigned int*)addr, __float_as_uint(v));
}

// ---------------- utility fill ----------------
__global__ void fill_kernel(float* p, int n, float v) {
  int i = blockIdx.x * blockDim.x + threadIdx.x;
  if (i < n) p[i] = v;
}

// ---------------- pad + transpose small B matrix: BT[n*Kp + k] = B[k*N + n] ----------------
__global__ void padT_kernel(const float* __restrict__ B, int K, int N,
                            float* __restrict__ BT, int Kp, int Np) {
  int id = blockIdx.x * blockDim.x + threadIdx.x;
  if (id >= Np * Kp) return;
  int n = id / Kp, k = id - n * Kp;
  BT[id] = (k < K && n < N) ? B[k * N + n] : 0.f;
}

// ---------------- preprocess: embedding gather -> x [N1,52], x2 [N2,52] (col 51 = 0) ----------------
__global__ void pre_kernel(const int* __restrict__ cat1, const float* __restrict__ num1,
                           const int* __restrict__ cat2, const float* __restrict__ num2,
                           const float* __restrict__ emb0, const float* __restrict__ emb1,
                           const float* __restrict__ emb2,
                           float* __restrict__ x, float* __restrict__ x2) {
  int node = blockIdx.x;
  int t = threadIdx.x;
  const int* cat; const float* num; float* out;
  if (node < NN1) { cat = cat1 + node * 3; num = num1 + node * 3; out = x + node * XS; }
  else { int n2 = node - NN1; cat = cat2 + n2 * 3; num = num2 + n2 * 3; out = x2 + n2 * XS; }
  float v;
  if      (t < 16) v = emb0[cat[0] * 16 + t];
  else if (t < 24) v = emb1[cat[1] * 8  + (t - 16)];
  else if (t < 48) v = emb2[cat[2] * 24 + (t - 24)];
  else if (t < 51) v = num[t - 48];
  else if (t < 52) v = 0.f;   // zero pad column
  else return;
  out[t] = v;
}

// ---------------- padded f32 WMMA GEMM: C[M,Np] = A[M,Kp] @ BT^T, no guards ----------------
// A: row stride lda (even), K = Kp (multiple of 4, zero-padded via BT rows)
// BT: [Np][Kp] transposed+padded B. 4 waves per block; wave w does n-tile blockIdx.y*4+w.
__global__ void wmma_gemm_padded(const float* __restrict__ A, int lda,
                                 const float* __restrict__ BT,
                                 float* __restrict__ C, int ldc,
                                 int Kp, int ntiles) {
  int lane = threadIdx.x & 31;
  int wave = threadIdx.x >> 5;
  int nt = blockIdx.y * 4 + wave;
  if (nt >= ntiles) return;                    // wave-uniform
  int mt = blockIdx.x;
  int m  = lane & 15;
  int kh = (lane >> 4) * 2;
  const float* arow = A  + (mt * 16 + m) * lda + kh;
  const float* brow = BT + (nt * 16 + m) * Kp  + kh;
  v8f c = {};
  for (int k = 0; k < Kp; k += 4) {
    v2f a = *(const v2f*)(arow + k);
    v2f b = *(const v2f*)(brow + k);
    c = __builtin_amdgcn_wmma_f32_16x16x4_f32(false, a, false, b, (short)0, c, false, false);
  }
  int half = (lane >> 4) << 3;
  float* crow = C + (mt * 16 + half) * ldc + nt * 16 + m;
  for (int v = 0; v < 8; ++v) crow[v * ldc] = c[v];
}

// ---------------- GAT1 per-node attention coefficients ----------------
__global__ void attn1_node_kernel(const float* __restrict__ h1,
                                  const float* __restrict__ aw_src, const float* __restrict__ aw_dst,
                                  float* __restrict__ asrc, float* __restrict__ adst) {
  int id = blockIdx.x * blockDim.x + threadIdx.x;
  if (id >= NN1 * HEADS) return;
  int n = id >> 2, h = id & 3;
  const float* hp = h1 + n * NP1 + h * FM1;
  float s = 0.f, d = 0.f;
  for (int c = 0; c < FM1; ++c) { float hv = hp[c]; s += hv * aw_src[h * FM1 + c]; d += hv * aw_dst[h * FM1 + c]; }
  asrc[id] = s; adst[id] = d;
}

// ---------------- edge softmax passes (generic over both graphs) ----------------
__global__ void edge_max_kernel(const int* __restrict__ e, int E, int Nn,
                                const float* __restrict__ asrc, const float* __restrict__ adst,
                                float* __restrict__ mx) {
  int id = blockIdx.x * blockDim.x + threadIdx.x;
  if (id >= (E + Nn) * HEADS) return;
  int ed = id >> 2, h = id & 3;
  int s = ed < E ? e[ed] : ed - E;
  int d = ed < E ? e[E + ed] : ed - E;
  float v = leaky02(asrc[s * 4 + h] + adst[d * 4 + h]);
  atomicMaxF(&mx[d * 4 + h], v);
}

__global__ void edge_sum_kernel(const int* __restrict__ e, int E, int Nn,
                                const float* __restrict__ asrc, const float* __restrict__ adst,
                                const float* __restrict__ mx, float* __restrict__ sm) {
  int id = blockIdx.x * blockDim.x + threadIdx.x;
  if (id >= (E + Nn) * HEADS) return;
  int ed = id >> 2, h = id & 3;
  int s = ed < E ? e[ed] : ed - E;
  int d = ed < E ? e[E + ed] : ed - E;
  float v = leaky02(asrc[s * 4 + h] + adst[d * 4 + h]);
  atomicAdd(&sm[d * 4 + h], __expf(v - mx[d * 4 + h]));
}

// accumulate sum_h alpha_h * h1[src,h,c] into acc[dst,c]  (graph 1)
__global__ void edge_acc1_kernel(const int* __restrict__ e,
                                 const float* __restrict__ asrc, const float* __restrict__ adst,
                                 const float* __restrict__ mx, const float* __restrict__ sm,
                                 const float* __restrict__ h1, float* __restrict__ acc) {
  __shared__ float w[HEADS];
  int ed = blockIdx.x;
  int s = ed < EE1 ? e[ed] : ed - EE1;
  int d = ed < EE1 ? e[EE1 + ed] : ed - EE1;
  int t = threadIdx.x;
  if (t < HEADS) {
    float ev = leaky02(asrc[s * 4 + t] + adst[d * 4 + t]);
    w[t] = __expf(ev - mx[d * 4 + t]) / (sm[d * 4 + t] + 1e-16f);
  }
  __syncthreads();
  if (t < FM1) {
    const float* hp = h1 + s * NP1;
    float val = w[0] * hp[t] + w[1] * hp[FM1 + t] + w[2] * hp[2 * FM1 + t] + w[3] * hp[3 * FM1 + t];
    atomicAdd(&acc[d * FM1 + t], val);
  }
}

// x1[n, 0:50] = acc/4 + bias ; x1[n, 50:52] = 0   (stride XS=52)
__global__ void fin_x1_kernel(const float* __restrict__ acc, const float* __restrict__ bias,
                              float* __restrict__ x1) {
  int id = blockIdx.x * blockDim.x + threadIdx.x;
  if (id >= NN1 * XS) return;
  int n = id / XS, c = id - n * XS;
  x1[id] = (c < FM1) ? (acc[n * FM1 + c] * 0.25f + bias[c]) : 0.f;
}

// ---------------- row norms q = ||u||^2 ----------------
__global__ void qnorm_kernel(const float* __restrict__ u1, const float* __restrict__ u2,
                             float* __restrict__ q1, float* __restrict__ q2) {
  int id = blockIdx.x * blockDim.x + threadIdx.x;
  if (id >= NN1 + NN2) return;
  const float* p; float* q;
  if (id < NN1) { p = u1 + id * HIDD; q = q1 + id; }
  else          { p = u2 + (id - NN1) * HIDD; q = q2 + (id - NN1); }
  float s = 0.f;
  for (int k = 0; k < HIDD; ++k) { float v = p[k]; s += v * v; }
  *q = s;
}

// ---------------- fused per-batch: C = u2@u1^T (WMMA), mask, softmax, alpha@y_past ----------------
__global__ void cross_softmax_kernel(const float* __restrict__ u1, const float* __restrict__ u2,
                                     const float* __restrict__ q1, const float* __restrict__ q2,
                                     const int* __restrict__ Amat, const float* __restrict__ x,
                                     float* __restrict__ tmp) {
  __shared__ float al[FUTN][PASTN];
  int b = blockIdx.x;
  int tid = threadIdx.x;
  int wave = tid >> 5, lane = tid & 31;
  const float* u1b = u1 + b * PASTN * HIDD;
  const float* u2b = u2 + b * FUTN * HIDD;
  const float* q1b = q1 + b * PASTN;
  const float* q2b = q2 + b * FUTN;
  int m = lane & 15;
  int kh = (lane >> 4) * 2;
  // 4 f-tiles x 8 p-tiles = 32 tiles across 8 waves
  for (int t = wave; t < 32; t += 8) {
    int ft = t >> 3, pt = t & 7;
    const float* arow = u2b + (ft * 16 + m) * HIDD + kh;
    const float* brow = u1b + (pt * 16 + m) * HIDD + kh;
    v8f c = {};
    for (int k = 0; k < HIDD; k += 4) {
      v2f a  = *(const v2f*)(arow + k);
      v2f bb = *(const v2f*)(brow + k);
      c = __builtin_amdgcn_wmma_f32_16x16x4_f32(false, a, false, bb, (short)0, c, false, false);
    }
    int half = (lane >> 4) << 3;
    for (int v = 0; v < 8; ++v) {
      int f = ft * 16 + v + half;
      int p = pt * 16 + m;
      float val = 2.f * c[v] - q1b[p] - q2b[f];
      if (Amat[p * ADIM + PASTN + f] == 0) val = -__builtin_inff();
      al[f][p] = val;
    }
  }
  __syncthreads();
  // masked softmax over p and dot with y_past = x[:, 50]
  for (int r = 0; r < 8; ++r) {
    int f = wave * 8 + r;
    float v0 = al[f][lane], v1 = al[f][lane + 32], v2 = al[f][lane + 64], v3 = al[f][lane + 96];
    float mx = fmaxf(fmaxf(v0, v1), fmaxf(v2, v3));
    for (int o = 16; o; o >>= 1) mx = fmaxf(mx, __shfl_xor(mx, o, 32));
    float y0 = x[(b * PASTN + lane) * XS + 50];
    float y1 = x[(b * PASTN + lane + 32) * XS + 50];
    float y2 = x[(b * PASTN + lane + 64) * XS + 50];
    float y3 = x[(b * PASTN + lane + 96) * XS + 50];
    float e0 = __expf(v0 - mx), e1 = __expf(v1 - mx), e2 = __expf(v2 - mx), e3 = __expf(v3 - mx);
    float se = e0 + e1 + e2 + e3;
    float sw = e0 * y0 + e1 * y1 + e2 * y2 + e3 * y3;
    for (int o = 16; o; o >>= 1) { se += __shfl_xor(se, o, 32); sw += __shfl_xor(sw, o, 32); }
    if (lane == 0) tmp[b * FUTN + f] = sw / se;
  }
}

// ---------------- GAT2 per-node: h2 / attention coeffs ----------------
__global__ void node2_kernel(const float* __restrict__ x2, const float* __restrict__ tmp,
                             const float* __restrict__ g2lin,
                             const float* __restrict__ g2as, const float* __restrict__ g2ad,
                             float* __restrict__ h2, float* __restrict__ as2, float* __restrict__ ad2) {
  int id = blockIdx.x * blockDim.x + threadIdx.x;
  if (id >= NN2 * HEADS) return;
  int n = id >> 2, h = id & 3;
  const float* xr = x2 + n * XS;
  float s = 0.f;
  for (int k = 0; k < FM1; ++k) s += xr[k] * g2lin[k * HEADS + h];
  s += tmp[n] * g2lin[FM1 * HEADS + h];
  h2[id] = s; as2[id] = s * g2as[h]; ad2[id] = s * g2ad[h];
}

__global__ void edge_acc2_kernel(const int* __restrict__ e,
                                 const float* __restrict__ as2, const float* __restrict__ ad2,
                                 const float* __restrict__ mx, const float* __restrict__ sm,
                                 const float* __restrict__ h2, float* __restrict__ acc) {
  int ed = blockIdx.x * blockDim.x + threadIdx.x;
  if (ed >= EE2 + NN2) return;
  int s = ed < EE2 ? e[ed] : ed - EE2;
  int d = ed < EE2 ? e[EE2 + ed] : ed - EE2;
  float val = 0.f;
  for (int h = 0; h < HEADS; ++h) {
    float ev = leaky02(as2[s * 4 + h] + ad2[d * 4 + h]);
    float w = __expf(ev - mx[d * 4 + h]) / (sm[d * 4 + h] + 1e-16f);
    val += w * h2[s * 4 + h];
  }
  atomicAdd(&acc[d], val);
}

__global__ void fin_out_kernel(const float* __restrict__ acc, const float* __restrict__ g2b,
                               float* __restrict__ out) {
  int id = blockIdx.x * blockDim.x + threadIdx.x;
  if (id >= NN2) return;
  out[id] = acc[id] * 0.25f + g2b[0];
}

// ---------------- host launch ----------------
static inline int cdiv(int a, int b) { return (a + b - 1) / b; }

extern "C" void kernel_launch(void* const* d_in, const int* in_sizes, int n_in,
                              void* d_out, int out_size, void* d_ws, size_t ws_size,
                              hipStream_t stream) {
  const int*   cat1   = (const int*)d_in[0];
  const float* num1   = (const float*)d_in[1];
  const int*   cat2   = (const int*)d_in[2];
  const float* num2   = (const float*)d_in[3];
  const int*   e1     = (const int*)d_in[4];
  const int*   e2     = (const int*)d_in[5];
  const int*   Amat   = (const int*)d_in[6];
  const float* emb0   = (const float*)d_in[7];
  const float* emb1   = (const float*)d_in[8];
  const float* emb2   = (const float*)d_in[9];
  const float* g1_lin = (const float*)d_in[10];
  const float* g1_as  = (const float*)d_in[11];
  const float* g1_ad  = (const float*)d_in[12];
  const float* g1_b   = (const float*)d_in[13];
  const float* g2_lin = (const float*)d_in[14];
  const float* g2_as  = (const float*)d_in[15];
  const float* g2_ad  = (const float*)d_in[16];
  const float* g2_b   = (const float*)d_in[17];
  const float* Wm     = (const float*)d_in[18];
  float* out = (float*)d_out;
  float* ws = (float*)d_ws;

  // workspace layout (floats; every block even-sized -> 8B alignment preserved)
  size_t o = 0;
  float* X    = ws + o; o += (size_t)NN1 * XS;          // [8192,52]
  float* X2   = ws + o; o += (size_t)NN2 * XS;          // [4096,52]
  float* H1   = ws + o; o += (size_t)NN1 * NP1;         // [8192,208]
  float* AS1  = ws + o; o += (size_t)NN1 * HEADS;
  float* AD1  = ws + o; o += (size_t)NN1 * HEADS;
  float* X1   = ws + o; o += (size_t)NN1 * XS;          // [8192,52]
  float* U1   = ws + o; o += (size_t)NN1 * HIDD;
  float* U2   = ws + o; o += (size_t)NN2 * HIDD;
  float* Q1   = ws + o; o += NN1;
  float* Q2   = ws + o; o += NN2;
  float* TMPB = ws + o; o += NN2;
  float* H2   = ws + o; o += (size_t)NN2 * HEADS;
  float* AS2  = ws + o; o += (size_t)NN2 * HEADS;
  float* AD2  = ws + o; o += (size_t)NN2 * HEADS;
  float* G1T  = ws + o; o += (size_t)NP1 * XS;          // [208,52] transposed+padded g1_lin
  float* WT   = ws + o; o += (size_t)HIDD * XS;         // [64,52]  transposed+padded W
  float* MAX1 = ws + o; o += (size_t)NN1 * HEADS;       // contiguous: MAX1, MAX2
  float* MAX2 = ws + o; o += (size_t)NN2 * HEADS;
  float* SUM1 = ws + o; o += (size_t)NN1 * HEADS;       // contiguous: SUM1, SUM2, ACC1, ACC2
  float* SUM2 = ws + o; o += (size_t)NN2 * HEADS;
  float* ACC1 = ws + o; o += (size_t)NN1 * FM1;
  float* ACC2 = ws + o; o += NN2;

  // init: -inf for max buffers, 0 for sum/acc buffers
  {
    int nInf = (NN1 + NN2) * HEADS;
    fill_kernel<<<cdiv(nInf, 256), 256, 0, stream>>>(MAX1, nInf, -__builtin_inff());
    int nZero = (NN1 + NN2) * HEADS + NN1 * FM1 + NN2;
    fill_kernel<<<cdiv(nZero, 256), 256, 0, stream>>>(SUM1, nZero, 0.f);
  }

  // 0) pad+transpose weight matrices for guard-free WMMA
  padT_kernel<<<cdiv(NP1 * XS, 256), 256, 0, stream>>>(g1_lin, FDIM, HEADS * FM1, G1T, XS, NP1);
  padT_kernel<<<cdiv(HIDD * XS, 256), 256, 0, stream>>>(Wm, FM1, HIDD, WT, XS, HIDD);

  // 1) preprocess (embedding gathers, zero-padded stride 52)
  pre_kernel<<<NN1 + NN2, 64, 0, stream>>>(cat1, num1, cat2, num2, emb0, emb1, emb2, X, X2);

  // 2) h1 = x @ g1_lin   [8192,52]x[52,208]  (f32 WMMA, guard-free)
  wmma_gemm_padded<<<dim3(NN1 / 16, cdiv(NP1 / 16, 4)), 128, 0, stream>>>(
      X, XS, G1T, H1, NP1, XS, NP1 / 16);

  // 3) per-node attention coefficients for GAT1
  attn1_node_kernel<<<cdiv(NN1 * HEADS, 256), 256, 0, stream>>>(H1, g1_as, g1_ad, AS1, AD1);

  // 4) edge softmax passes for GAT1 (edges + self loops)
  {
    int tot = (EE1 + NN1) * HEADS;
    edge_max_kernel<<<cdiv(tot, 256), 256, 0, stream>>>(e1, EE1, NN1, AS1, AD1, MAX1);
    edge_sum_kernel<<<cdiv(tot, 256), 256, 0, stream>>>(e1, EE1, NN1, AS1, AD1, MAX1, SUM1);
    edge_acc1_kernel<<<EE1 + NN1, 64, 0, stream>>>(e1, AS1, AD1, MAX1, SUM1, H1, ACC1);
  }

  // 5) x1 = head-mean + bias (stride 52, zero tail)
  fin_x1_kernel<<<cdiv(NN1 * XS, 256), 256, 0, stream>>>(ACC1, g1_b, X1);

  // 6) u1 = x1 @ W ; u2 = x2[:, :50] @ W   (zero rows of WT nullify pad cols)
  wmma_gemm_padded<<<dim3(NN1 / 16, 1), 128, 0, stream>>>(X1, XS, WT, U1, HIDD, XS, HIDD / 16);
  wmma_gemm_padded<<<dim3(NN2 / 16, 1), 128, 0, stream>>>(X2, XS, WT, U2, HIDD, XS, HIDD / 16);

  // 7) row norms
  qnorm_kernel<<<cdiv(NN1 + NN2, 256), 256, 0, stream>>>(U1, U2, Q1, Q2);

  // 8) fused per-batch: cross WMMA GEMM + mask + softmax + alpha@y_past
  cross_softmax_kernel<<<NB, 256, 0, stream>>>(U1, U2, Q1, Q2, Amat, X, TMPB);

  // 9) GAT2 per-node features + attention coefficients
  node2_kernel<<<cdiv(NN2 * HEADS, 256), 256, 0, stream>>>(X2, TMPB, g2_lin, g2_as, g2_ad, H2, AS2, AD2);

  // 10) edge softmax passes for GAT2
  {
    int tot = (EE2 + NN2) * HEADS;
    edge_max_kernel<<<cdiv(tot, 256), 256, 0, stream>>>(e2, EE2, NN2, AS2, AD2, MAX2);
    edge_sum_kernel<<<cdiv(tot, 256), 256, 0, stream>>>(e2, EE2, NN2, AS2, AD2, MAX2, SUM2);
    edge_acc2_kernel<<<cdiv(EE2 + NN2, 256), 256, 0, stream>>>(e2, AS2, AD2, MAX2, SUM2, H2, ACC2);
  }

  // 11) final output
  fin_out_kernel<<<cdiv(NN2, 256), 256, 0, stream>>>(ACC2, g2_b, out);

  (void)in_sizes; (void)n_in; (void)out_size; (void)ws_size;
}